// EpropGateL0rd_17927193494438
// MI455X (gfx1250) — compile-verified
//
#include <hip/hip_runtime.h>

typedef __attribute__((ext_vector_type(2))) float v2f;
typedef __attribute__((ext_vector_type(4))) float v4f;
typedef __attribute__((ext_vector_type(8))) float v8f;

#define B_  128
#define NI_ 512
#define NH_ 512
#define NO_ 512

// d_out offsets (in floats), concatenated in reference return order
#define O_OUT   0
#define O_H     65536
#define O_EWGX  131072
#define O_EWGH  33685504
#define O_EBG   67239936
#define O_EWRX  67305472
#define O_EWRH  100859904
#define O_EBR   134414336
#define O_OPEN  134479872

// workspace offsets (in floats)
#define W_ACC 0
#define W_C0  16
#define W_C1G (16 + 65536)
#define W_C1R (16 + 131072)

// Load a 16x4 (A) or 4x16-transposed (B) fp32 WMMA fragment.
// Per ISA 7.12.2: VGPR0 holds K=0 (lanes 0-15) / K=2 (lanes 16-31),
// VGPR1 holds K=1 / K=3 -> per lane this is a contiguous float2.
__device__ __forceinline__ v2f load_frag(const float* __restrict__ p, int row0,
                                         int ld, int k, int lane) {
    int r = row0 + (lane & 15);
    int c = k + ((lane >> 4) << 1);
    return *(const v2f*)(p + (size_t)r * ld + c);
}

__device__ __forceinline__ float sigmoidf_(float v) {
    return 1.0f / (1.0f + __expf(-v));
}

// ---------------- gate kernel: g, r, h, coefficients, e_b_* , openings ----
__global__ __launch_bounds__(256) void gate_kernel(
    const float* __restrict__ x, const float* __restrict__ h_last,
    const float* __restrict__ w_gx, const float* __restrict__ w_gh,
    const float* __restrict__ b_g,
    const float* __restrict__ w_rx, const float* __restrict__ w_rh,
    const float* __restrict__ b_r,
    const float* __restrict__ e_b_g, const float* __restrict__ e_b_r,
    float* __restrict__ out_base, float* __restrict__ ws)
{
    const int tid  = threadIdx.x;
    const int lane = tid & 31;
    const int wave = tid >> 5;
    const int gt   = blockIdx.x * 8 + wave;   // 256 tiles total
    const int m0   = (gt & 7) * 16;           // B dimension
    const int n0   = (gt >> 3) * 16;          // NH dimension

    v8f accg = {};
    v8f accr = {};
    for (int k = 0; k < NI_; k += 4) {
        v2f ax  = load_frag(x,      m0, NI_, k, lane);
        v2f ah  = load_frag(h_last, m0, NH_, k, lane);
        v2f bgx = load_frag(w_gx,   n0, NI_, k, lane);
        v2f bgh = load_frag(w_gh,   n0, NH_, k, lane);
        v2f brx = load_frag(w_rx,   n0, NI_, k, lane);
        v2f brh = load_frag(w_rh,   n0, NH_, k, lane);
        accg = __builtin_amdgcn_wmma_f32_16x16x4_f32(false, ax, false, bgx, (short)0, accg, false, false);
        accg = __builtin_amdgcn_wmma_f32_16x16x4_f32(false, ah, false, bgh, (short)0, accg, false, false);
        accr = __builtin_amdgcn_wmma_f32_16x16x4_f32(false, ax, false, brx, (short)0, accr, false, false);
        accr = __builtin_amdgcn_wmma_f32_16x16x4_f32(false, ah, false, brh, (short)0, accr, false, false);
    }

    const int half = lane >> 4;
    const int j    = n0 + (lane & 15);
    const float bg = b_g[j];
    const float br = b_r[j];
    float hsum = 0.0f;

#pragma unroll
    for (int v = 0; v < 8; ++v) {
        const int b      = m0 + v + half * 8;       // C layout: M = v (+8 for hi half)
        const int rowIdx = b * NH_ + j;
        const float ag = accg[v] + bg;
        const float ar = accr[v] + br;
        float g = tanhf(ag);
        g = fmaxf(g, 0.0f);
        const float r  = tanhf(ar);
        const float hl = h_last[rowIdx];
        const float h  = g * r + (1.0f - g) * hl;
        const float Hg = (g > 0.0f) ? 1.0f : 0.0f;
        const float dg = (1.0f - g * g) * Hg;
        const float dr = 1.0f - r * r;
        const float c0  = 1.0f - g;
        const float c1g = dg * (r - hl);
        const float c1r = dr * g;

        out_base[O_H + rowIdx] = h;
        ws[W_C0  + rowIdx] = c0;
        ws[W_C1G + rowIdx] = c1g;
        ws[W_C1R + rowIdx] = c1r;
        out_base[O_EBG + rowIdx] = e_b_g[rowIdx] * c0 + c1g;
        out_base[O_EBR + rowIdx] = e_b_r[rowIdx] * c0 + c1r;
        hsum += Hg;
    }

    // wave32 reduction (integer-valued floats -> exact, order independent)
    for (int off = 16; off > 0; off >>= 1)
        hsum += __shfl_down(hsum, off, 32);
    if (lane == 0)
        atomicAdd(ws + W_ACC, hsum);
}

// ---------------- trace kernel: the ~1 GB streaming update ----------------
__global__ __launch_bounds__(256) void trace_kernel(
    const float* __restrict__ x, const float* __restrict__ h_last,
    const float* __restrict__ e_w_gx, const float* __restrict__ e_w_gh,
    const float* __restrict__ e_w_rx, const float* __restrict__ e_w_rh,
    const float* __restrict__ ws, float* __restrict__ out_base)
{
    const int idx    = blockIdx.x * 256 + threadIdx.x;  // exactly B*NH*NI/4 threads
    const int chunk  = idx & 127;                       // float4 chunk within row
    const int rowIdx = idx >> 7;                        // b*NH + j
    const int b      = rowIdx >> 9;                     // NH == 512

    const v4f x4 = ((const v4f*)x)[(size_t)b * 128 + chunk];
    const v4f h4 = ((const v4f*)h_last)[(size_t)b * 128 + chunk];
    const float c0  = ws[W_C0  + rowIdx];
    const float c1g = ws[W_C1G + rowIdx];
    const float c1r = ws[W_C1R + rowIdx];
    const size_t e4 = (size_t)rowIdx * 128 + chunk;

    // streaming tensors: non-temporal so 1 GB doesn't evict the reused data
    v4f egx = __builtin_nontemporal_load((const v4f*)e_w_gx + e4);
    __builtin_nontemporal_store(egx * c0 + x4 * c1g, (v4f*)(out_base + O_EWGX) + e4);
    v4f egh = __builtin_nontemporal_load((const v4f*)e_w_gh + e4);
    __builtin_nontemporal_store(egh * c0 + h4 * c1g, (v4f*)(out_base + O_EWGH) + e4);
    v4f erx = __builtin_nontemporal_load((const v4f*)e_w_rx + e4);
    __builtin_nontemporal_store(erx * c0 + x4 * c1r, (v4f*)(out_base + O_EWRX) + e4);
    v4f erh = __builtin_nontemporal_load((const v4f*)e_w_rh + e4);
    __builtin_nontemporal_store(erh * c0 + h4 * c1r, (v4f*)(out_base + O_EWRH) + e4);
}

// ---------------- output kernel: p, o -> out = sigmoid(o)*tanh(p) ---------
__global__ __launch_bounds__(256) void out_kernel(
    const float* __restrict__ x, const float* __restrict__ h,
    const float* __restrict__ w_px, const float* __restrict__ w_ph,
    const float* __restrict__ b_p,
    const float* __restrict__ w_ox, const float* __restrict__ w_oh,
    const float* __restrict__ b_o,
    float* __restrict__ out_base)
{
    const int tid  = threadIdx.x;
    const int lane = tid & 31;
    const int wave = tid >> 5;
    const int gt   = blockIdx.x * 8 + wave;
    const int m0   = (gt & 7) * 16;           // B dimension
    const int n0   = (gt >> 3) * 16;          // NO dimension

    v8f accp = {};
    v8f acco = {};
    for (int k = 0; k < NI_; k += 4) {
        v2f ax  = load_frag(x,    m0, NI_, k, lane);
        v2f ah  = load_frag(h,    m0, NH_, k, lane);
        v2f bpx = load_frag(w_px, n0, NI_, k, lane);
        v2f bph = load_frag(w_ph, n0, NH_, k, lane);
        v2f box = load_frag(w_ox, n0, NI_, k, lane);
        v2f boh = load_frag(w_oh, n0, NH_, k, lane);
        accp = __builtin_amdgcn_wmma_f32_16x16x4_f32(false, ax, false, bpx, (short)0, accp, false, false);
        accp = __builtin_amdgcn_wmma_f32_16x16x4_f32(false, ah, false, bph, (short)0, accp, false, false);
        acco = __builtin_amdgcn_wmma_f32_16x16x4_f32(false, ax, false, box, (short)0, acco, false, false);
        acco = __builtin_amdgcn_wmma_f32_16x16x4_f32(false, ah, false, boh, (short)0, acco, false, false);
    }

    const int half = lane >> 4;
    const int j    = n0 + (lane & 15);
    const float bp = b_p[j];
    const float bo = b_o[j];

#pragma unroll
    for (int v = 0; v < 8; ++v) {
        const int b  = m0 + v + half * 8;
        const float p = tanhf(accp[v] + bp);
        const float o = sigmoidf_(acco[v] + bo);
        out_base[O_OUT + b * NO_ + j] = o * p;
    }
}

// ---------------- finalize: openings ---------------------------------------
__global__ void finalize_kernel(const float* __restrict__ ws,
                                float* __restrict__ out_base) {
    out_base[O_OPEN] = ws[W_ACC] * (1.0f / 65536.0f);
}

extern "C" void kernel_launch(void* const* d_in, const int* in_sizes, int n_in,
                              void* d_out, int out_size, void* d_ws, size_t ws_size,
                              hipStream_t stream) {
    const float* x      = (const float*)d_in[0];
    const float* h_last = (const float*)d_in[1];
    const float* w_gx   = (const float*)d_in[2];
    const float* w_gh   = (const float*)d_in[3];
    const float* b_g    = (const float*)d_in[4];
    const float* w_rx   = (const float*)d_in[5];
    const float* w_rh   = (const float*)d_in[6];
    const float* b_r    = (const float*)d_in[7];
    const float* w_px   = (const float*)d_in[8];
    const float* w_ph   = (const float*)d_in[9];
    const float* b_p    = (const float*)d_in[10];
    const float* w_ox   = (const float*)d_in[11];
    const float* w_oh   = (const float*)d_in[12];
    const float* b_o    = (const float*)d_in[13];
    const float* e_w_gx = (const float*)d_in[14];
    const float* e_w_gh = (const float*)d_in[15];
    const float* e_b_g  = (const float*)d_in[16];
    const float* e_w_rx = (const float*)d_in[17];
    const float* e_w_rh = (const float*)d_in[18];
    const float* e_b_r  = (const float*)d_in[19];
    float* out = (float*)d_out;
    float* ws  = (float*)d_ws;

    hipMemsetAsync(d_ws, 0, 64, stream);  // zero openings accumulator (capture-safe)

    gate_kernel<<<32, 256, 0, stream>>>(x, h_last, w_gx, w_gh, b_g,
                                        w_rx, w_rh, b_r, e_b_g, e_b_r, out, ws);
    trace_kernel<<<32768, 256, 0, stream>>>(x, h_last, e_w_gx, e_w_gh,
                                            e_w_rx, e_w_rh, ws, out);
    out_kernel<<<32, 256, 0, stream>>>(x, out + O_H, w_px, w_ph, b_p,
                                       w_ox, w_oh, b_o, out);
    finalize_kernel<<<1, 1, 0, stream>>>(ws, out);
}